// CNNEmbedder_2843268350681
// MI455X (gfx1250) — compile-verified
//
#include <hip/hip_runtime.h>

#define A_DIM    48
#define R_ROUNDS 25
#define T_STEPS  23          // R - 2
#define NPAIR    1128        // off-diagonal upper-tri pairs
#define ND_TOT   1176        // 48 diag + 1128 nondiag
#define BATCH    4096
#define BLK      128

typedef float __attribute__((ext_vector_type(4))) vfloat4;

__device__ __forceinline__ float sigmoidf(float x) {
    return 1.0f / (1.0f + __expf(-x));
}

// ---------------------------------------------------------------------------
// Prep: sigmoid tables + triangular pair decode into workspace.
//   d_ws layout: float sig[48] | float fnd[1128*4] | u16 pairs[1128]
// ---------------------------------------------------------------------------
__global__ void prep_kernel(const float* __restrict__ emb_diag,
                            const float* __restrict__ emb_nd,
                            float* __restrict__ w_sig,
                            float* __restrict__ w_fnd,
                            unsigned short* __restrict__ w_pairs) {
    int g = blockIdx.x * blockDim.x + threadIdx.x;
    if (g < A_DIM) {
        w_sig[g] = sigmoidf(emb_diag[g]);
    }
    if (g < NPAIR) {
        float f12 = sigmoidf(emb_nd[g * 4 + 0]);
        float f9  = sigmoidf(emb_nd[g * 4 + 1]) * f12;
        float f8  = sigmoidf(emb_nd[g * 4 + 2]) * f9;
        float f6  = sigmoidf(emb_nd[g * 4 + 3]) * f8;
        w_fnd[g * 4 + 0] = f12;
        w_fnd[g * 4 + 1] = f9;
        w_fnd[g * 4 + 2] = f8;
        w_fnd[g * 4 + 3] = f6;
        // decode q -> (iy, ix), iy < ix, row-major over iy
        int q = g, iy = 0;
        while (q >= (A_DIM - 1) - iy) { q -= (A_DIM - 1) - iy; ++iy; }
        int ix = iy + 1 + q;
        w_pairs[g] = (unsigned short)((iy << 8) | ix);
    }
}

// ---------------------------------------------------------------------------
// Async global -> LDS staging (CDNA5 async path, ASYNCcnt-tracked).
// bytes must be a multiple of 16; both pointers 16B aligned.
// ---------------------------------------------------------------------------
__device__ __forceinline__ void stage_async16(void* lds_dst, const void* gsrc,
                                              int bytes, int tid, int nthr) {
    unsigned int lds0 = (unsigned int)(unsigned long long)(uintptr_t)lds_dst;
    const char* gp = (const char*)gsrc;
    int n = bytes >> 4;
    for (int i = tid; i < n; i += nthr) {
        unsigned int lo = lds0 + ((unsigned int)i << 4);
        const void* ga = gp + ((size_t)i << 4);
        asm volatile("global_load_async_to_lds_b128 %0, %1, off"
                     :: "v"(lo), "v"(ga) : "memory");
    }
}

// ---------------------------------------------------------------------------
// Main kernel: one batch element per block.
// ---------------------------------------------------------------------------
__global__ __launch_bounds__(BLK)
void syndrome_stream_kernel(const int* __restrict__ x,
                            const float* __restrict__ w_sig,
                            const float* __restrict__ w_fnd,
                            const unsigned short* __restrict__ w_pairs,
                            float* __restrict__ out) {
    __shared__ alignas(16) int            s_x[R_ROUNDS * A_DIM]; // 4800 B
    __shared__ alignas(16) float          s_sig[A_DIM];          // 192 B
    __shared__ alignas(16) float          s_fnd[NPAIR * 4];      // 18048 B
    __shared__ alignas(16) unsigned short s_pairs[NPAIR];        // 2256 B
    __shared__ int                        s_xx[A_DIM];

    const int tid = threadIdx.x;
    const int b   = blockIdx.x;

    stage_async16(s_x,    x + (size_t)b * (R_ROUNDS * A_DIM),
                  R_ROUNDS * A_DIM * 4, tid, BLK);
    stage_async16(s_fnd,  w_fnd,   NPAIR * 16, tid, BLK);
    stage_async16(s_sig,  w_sig,   A_DIM * 4,  tid, BLK);
    stage_async16(s_pairs, w_pairs, NPAIR * 2, tid, BLK);
    asm volatile("s_wait_asynccnt 0x0" ::: "memory");
    __syncthreads();

    int st = -1, dt = 1;   // exact int recurrence state (lanes 0..47)

    for (int t = 0; t < T_STEPS; ++t) {
        if (tid < A_DIM) {
            int ai = s_x[t * A_DIM + tid];
            int bi = s_x[(t + 1) * A_DIM + tid];
            int ci = s_x[(t + 2) * A_DIM + tid];
            int de = ai + ci - 2 * ai * ci;                 // data_err
            int me = bi * (1 - ai - ci) + ai * ci;          // meas_err
            dt = dt * (1 - 2 * me);
            int s = st + dt * de;
            st = (s < -1) ? -1 : ((s > 1) ? 1 : s);         // clip
            int ome = 1 - me;
            dt = dt * (1 - st * st * ome) - st * ome;
            s_xx[tid] = st + 3;                             // in {2,3,4}
        }
        __syncthreads();

        float* orow = out + ((size_t)b * T_STEPS + t) * ND_TOT;
        for (int v = tid; v < ND_TOT / 4; v += BLK) {
            vfloat4 r;
            if (v < A_DIM / 4) {
                // diagonal block: xx^2 in {4,9,16} -> {0, sig, 1}
                int p = v * 4;
                #pragma unroll
                for (int k = 0; k < 4; ++k) {
                    int xv = s_xx[p + k];
                    r[k] = (xv == 4) ? 1.0f : ((xv == 3) ? s_sig[p + k] : 0.0f);
                }
            } else {
                int q0 = v * 4 - A_DIM;
                #pragma unroll
                for (int k = 0; k < 4; ++k) {
                    int q = q0 + k;
                    unsigned pr = s_pairs[q];
                    int prod = s_xx[pr >> 8] * s_xx[pr & 255];
                    float val;
                    if (prod == 16)      val = 1.0f;
                    else if (prod == 4)  val = 0.0f;
                    else {
                        int sel = (prod == 12) ? 0 : (prod == 9) ? 1
                                : (prod == 8) ? 2 : 3;
                        val = s_fnd[q * 4 + sel];
                    }
                    r[k] = val;
                }
            }
            // 443 MB output stream >> 192 MB L2: bypass with non-temporal b128
            __builtin_nontemporal_store(r, (vfloat4*)orow + v);
        }
        __syncthreads();   // s_xx rewritten next step
    }
}

// ---------------------------------------------------------------------------
extern "C" void kernel_launch(void* const* d_in, const int* in_sizes, int n_in,
                              void* d_out, int out_size, void* d_ws, size_t ws_size,
                              hipStream_t stream) {
    const int*   x  = (const int*)d_in[0];       // (4096, 25, 48) int32
    const float* ed = (const float*)d_in[1];     // (1, 48) f32
    const float* en = (const float*)d_in[2];     // (1, 1128, 4) f32

    float* w_sig = (float*)d_ws;
    float* w_fnd = w_sig + A_DIM;
    unsigned short* w_pairs =
        (unsigned short*)((char*)d_ws + (size_t)(A_DIM + NPAIR * 4) * sizeof(float));

    prep_kernel<<<(NPAIR + 255) / 256, 256, 0, stream>>>(ed, en, w_sig, w_fnd, w_pairs);
    syndrome_stream_kernel<<<BATCH, BLK, 0, stream>>>(x, w_sig, w_fnd, w_pairs,
                                                      (float*)d_out);
}